// MambaChallenge2Model_71829033059149
// MI455X (gfx1250) — compile-verified
//
#include <hip/hip_runtime.h>
#include <hip/hip_bf16.h>
#include <math.h>

#define EPS 1e-5f

typedef __attribute__((ext_vector_type(2))) float v2f;
typedef __attribute__((ext_vector_type(8))) float v8f;

// ---------------- model dims ----------------
constexpr int BATCH   = 4;
constexpr int NWIN    = 16;
constexpr int BW      = BATCH * NWIN;   // 64
constexpr int CIN     = 129;
constexpr int T       = 200;
constexpr int BT      = BW * T;         // 12800
constexpr int DM      = 256;            // d_model
constexpr int DI      = 512;            // d_inner
constexpr int DS      = 32;             // d_state
constexpr int NH      = 16;
constexpr int HD      = 32;
constexpr int CONVD   = DI + 2 * DS;    // 576
constexpr int DIP     = 2*DI + 2*DS + NH; // 1104
constexpr int NLAYERS = 6;
constexpr int NC      = 4;              // chunks
constexpr int CH      = 50;             // chunk len

__device__ __forceinline__ float geluf(float x) {
  return 0.5f * x * (1.0f + erff(x * 0.7071067811865475f));
}
__device__ __forceinline__ float siluf(float x) {
  return x / (1.0f + expf(-x));
}
__device__ __forceinline__ float softplusf(float x) {
  return (x > 20.0f) ? x : log1pf(expf(x));
}

// ======================================================================
// WMMA fp32 GEMM:  C[M,N] = A[M,K] @ W[N,K]^T (+ bias[N])
// One wave computes a 16x64 tile (4 accumulators) with
// V_WMMA_F32_16X16X4_F32, sharing one A fragment across 4 WMMAs per
// k-step: 8B A + 4*8B B per lane per 4 WMMAs (~6.5 FLOP/B).
// Tile coordinates are pushed through readfirstlane so ALL control flow
// is scalar (s_cmp/s_cbranch) and EXEC provably stays all-ones for every
// WMMA, as the ISA requires.
// fp32 A-fragment layout (16x4): lanes 0-15 hold K=0,1; lanes 16-31 K=2,3
// -> each lane loads one contiguous float2 from its row per k-step.
// C-fragment: VGPR r -> M = r (+8 for upper half-wave), N = lane&15.
// Requires: M,N multiples of 16, K multiple of 4 (true for every call).
// ======================================================================
__global__ void gemm_nt_wmma(const float* __restrict__ A, const float* __restrict__ W,
                             const float* __restrict__ bias, float* __restrict__ C,
                             int M, int N, int K) {
  const int wavesPerBlock = blockDim.x >> 5;
  // wave index is uniform across the wave: make it an SGPR value so the
  // compiler emits scalar branches (no EXEC masking around WMMA).
  int wave = __builtin_amdgcn_readfirstlane(
      blockIdx.x * wavesPerBlock + (threadIdx.x >> 5));
  int ntiles  = N >> 4;
  int ngroups = (ntiles + 3) >> 2;
  int totalWaves = (M >> 4) * ngroups;
  if (wave >= totalWaves) return;
  int mt  = wave / ngroups;
  int ng  = wave - mt * ngroups;
  int nt0 = ng << 2;
  int numNT = ntiles - nt0; if (numNT > 4) numNT = 4;   // SGPR-uniform

  int lane  = threadIdx.x & 31;
  int r15   = lane & 15;
  int khalf = (lane >> 4) << 1;             // 0 or 2
  const float* Ap  = A + (size_t)(mt * 16 + r15) * K + khalf;
  const float* Wp0 = W + (size_t)(nt0 * 16 + r15) * K + khalf;

  v8f acc0 = {}, acc1 = {}, acc2 = {}, acc3 = {};
  if (numNT == 4) {
    const float* Wp1 = Wp0 + (size_t)16 * K;
    const float* Wp2 = Wp0 + (size_t)32 * K;
    const float* Wp3 = Wp0 + (size_t)48 * K;
    for (int k = 0; k < K; k += 4) {
      v2f a  = *(const v2f*)(Ap  + k);
      v2f b0 = *(const v2f*)(Wp0 + k);
      v2f b1 = *(const v2f*)(Wp1 + k);
      v2f b2 = *(const v2f*)(Wp2 + k);
      v2f b3 = *(const v2f*)(Wp3 + k);
      acc0 = __builtin_amdgcn_wmma_f32_16x16x4_f32(false, a, false, b0, (short)0, acc0, false, false);
      acc1 = __builtin_amdgcn_wmma_f32_16x16x4_f32(false, a, false, b1, (short)0, acc1, false, false);
      acc2 = __builtin_amdgcn_wmma_f32_16x16x4_f32(false, a, false, b2, (short)0, acc2, false, false);
      acc3 = __builtin_amdgcn_wmma_f32_16x16x4_f32(false, a, false, b3, (short)0, acc3, false, false);
    }
  } else if (numNT == 3) {
    const float* Wp1 = Wp0 + (size_t)16 * K;
    const float* Wp2 = Wp0 + (size_t)32 * K;
    for (int k = 0; k < K; k += 4) {
      v2f a  = *(const v2f*)(Ap  + k);
      v2f b0 = *(const v2f*)(Wp0 + k);
      v2f b1 = *(const v2f*)(Wp1 + k);
      v2f b2 = *(const v2f*)(Wp2 + k);
      acc0 = __builtin_amdgcn_wmma_f32_16x16x4_f32(false, a, false, b0, (short)0, acc0, false, false);
      acc1 = __builtin_amdgcn_wmma_f32_16x16x4_f32(false, a, false, b1, (short)0, acc1, false, false);
      acc2 = __builtin_amdgcn_wmma_f32_16x16x4_f32(false, a, false, b2, (short)0, acc2, false, false);
    }
  } else if (numNT == 2) {
    const float* Wp1 = Wp0 + (size_t)16 * K;
    for (int k = 0; k < K; k += 4) {
      v2f a  = *(const v2f*)(Ap  + k);
      v2f b0 = *(const v2f*)(Wp0 + k);
      v2f b1 = *(const v2f*)(Wp1 + k);
      acc0 = __builtin_amdgcn_wmma_f32_16x16x4_f32(false, a, false, b0, (short)0, acc0, false, false);
      acc1 = __builtin_amdgcn_wmma_f32_16x16x4_f32(false, a, false, b1, (short)0, acc1, false, false);
    }
  } else {
    for (int k = 0; k < K; k += 4) {
      v2f a  = *(const v2f*)(Ap  + k);
      v2f b0 = *(const v2f*)(Wp0 + k);
      acc0 = __builtin_amdgcn_wmma_f32_16x16x4_f32(false, a, false, b0, (short)0, acc0, false, false);
    }
  }

  int mBase = mt * 16 + ((lane >> 4) << 3);
  v8f accs[4] = {acc0, acc1, acc2, acc3};
#pragma unroll
  for (int j = 0; j < 4; ++j) {
    if (j >= numNT) break;                  // scalar branch
    int col = (nt0 + j) * 16 + r15;
    float bv = bias ? bias[col] : 0.0f;
#pragma unroll
    for (int r = 0; r < 8; ++r)
      C[(size_t)(mBase + r) * N + col] = accs[j][r] + bv;
  }
}

// ======================================================================
// Front-end: conv1 (k=3, pad 1) + BN + GELU, output layout (b*t, 128)
// ======================================================================
__global__ void conv1_bn_gelu(const float* __restrict__ x, const float* __restrict__ w,
                              const float* __restrict__ cb,
                              const float* __restrict__ g, const float* __restrict__ bb,
                              const float* __restrict__ rm, const float* __restrict__ rv,
                              float* __restrict__ out) {
  int idx = blockIdx.x * blockDim.x + threadIdx.x;
  if (idx >= BW * 128 * T) return;
  int t = idx % T;
  int o = (idx / T) % 128;
  int b = idx / (T * 128);
  const float* xb = x + (size_t)b * CIN * T;
  const float* wo = w + (size_t)o * CIN * 3;
  float s = cb[o];
  for (int c = 0; c < CIN; ++c) {
    const float* xc = xb + (size_t)c * T;
    const float* wc = wo + c * 3;
    if (t >= 1)     s += wc[0] * xc[t - 1];
    s += wc[1] * xc[t];
    if (t <= T - 2) s += wc[2] * xc[t + 1];
  }
  s = (s - rm[o]) * rsqrtf(rv[o] + EPS) * g[o] + bb[o];
  out[((size_t)b * T + t) * 128 + o] = geluf(s);
}

// ======================================================================
// Mamba pieces
// ======================================================================
// depthwise causal conv (k=4, pad left 3) + SiLU on columns [512:1088) of zx
__global__ void dwconv_silu(const float* __restrict__ zx, const float* __restrict__ cw,
                            const float* __restrict__ cb, float* __restrict__ xbcc) {
  int idx = blockIdx.x * blockDim.x + threadIdx.x;
  if (idx >= BT * CONVD) return;
  int ch = idx % CONVD;
  int bt = idx / CONVD;
  int t  = bt % T;
  const float* w = cw + ch * 4;
  float s = cb[ch];
#pragma unroll
  for (int k = 0; k < 4; ++k) {
    int dt_ = t - (3 - k);
    if (dt_ >= 0) s += w[k] * zx[(size_t)(bt - (3 - k)) * DIP + DI + ch];
  }
  xbcc[(size_t)bt * CONVD + ch] = siluf(s);
}

__global__ void dt_softplus(const float* __restrict__ zx, const float* __restrict__ dtb,
                            float* __restrict__ dt) {
  int idx = blockIdx.x * blockDim.x + threadIdx.x;
  if (idx >= BT * NH) return;
  int h = idx & (NH - 1);
  int bt = idx >> 4;
  dt[idx] = softplusf(zx[(size_t)bt * DIP + DI + CONVD + h] + dtb[h]);
}

// per-(b,h,c): cumsum of dt*A over chunk; store A_cs and chunk decay
__global__ void ssd_cumsum(const float* __restrict__ dt, const float* __restrict__ Alog,
                           float* __restrict__ acs, float* __restrict__ cdec) {
  int idx = blockIdx.x * blockDim.x + threadIdx.x;
  if (idx >= BW * NH * NC) return;
  int c = idx % NC;
  int h = (idx / NC) % NH;
  int b = idx / (NC * NH);
  float A = -expf(Alog[h]);
  float acc = 0.0f;
  size_t obase = (((size_t)b * NH + h) * NC + c) * CH;
  for (int l = 0; l < CH; ++l) {
    int t = c * CH + l;
    acc += dt[((size_t)b * T + t) * NH + h] * A;
    acs[obase + l] = acc;
  }
  cdec[((size_t)b * NH + h) * NC + c] = expf(acc);
}

// per-(b,c,h): diagonal-block output Yd and end-of-chunk states
__global__ void ssd_chunk(const float* __restrict__ xbcc, const float* __restrict__ dt,
                          const float* __restrict__ acs,
                          float* __restrict__ yb, float* __restrict__ states) {
  int h = blockIdx.x % NH;
  int c = (blockIdx.x / NH) % NC;
  int b = blockIdx.x / (NH * NC);
  __shared__ float sB[CH][DS], sC[CH][DS], sX[CH][DS];
  __shared__ float sA[CH], sD[CH], sS[CH * CH];
  int tid = threadIdx.x;
  for (int i = tid; i < CH * DS; i += 256) {
    int l = i >> 5, n = i & 31;
    int t = c * CH + l;
    size_t base = ((size_t)b * T + t) * CONVD;
    sB[l][n] = xbcc[base + DI + n];
    sC[l][n] = xbcc[base + DI + DS + n];
    sX[l][n] = xbcc[base + h * HD + n] * dt[((size_t)b * T + t) * NH + h];
  }
  for (int l = tid; l < CH; l += 256)
    sA[l] = acs[(((size_t)b * NH + h) * NC + c) * CH + l];
  __syncthreads();
  float alast = sA[CH - 1];
  for (int l = tid; l < CH; l += 256) sD[l] = expf(alast - sA[l]);
  // masked decay-weighted scores S[l][s] = (C_l . B_s) * exp(acs_l - acs_s), s<=l
  for (int i = tid; i < CH * CH; i += 256) {
    int l = i / CH, s = i - l * CH;
    float v = 0.0f;
    if (s <= l) {
      float d = 0.0f;
      for (int n = 0; n < DS; ++n) d += sC[l][n] * sB[s][n];
      v = d * expf(sA[l] - sA[s]);
    }
    sS[i] = v;
  }
  __syncthreads();
  // Yd[l][p] = sum_{s<=l} S[l][s] * X[s][p]
  for (int i = tid; i < CH * HD; i += 256) {
    int l = i >> 5, p = i & 31;
    float acc = 0.0f;
    for (int s = 0; s <= l; ++s) acc += sS[l * CH + s] * sX[s][p];
    int t = c * CH + l;
    yb[((size_t)b * T + t) * DI + h * HD + p] = acc;
  }
  __syncthreads();
  // states[p][n] = sum_l B[l][n] * exp(acs_last - acs_l) * X[l][p]
  for (int i = tid; i < HD * DS; i += 256) {
    int p = i >> 5, n = i & 31;
    float acc = 0.0f;
    for (int l = 0; l < CH; ++l) acc += sB[l][n] * sD[l] * sX[l][p];
    states[((((size_t)b * NC + c) * NH + h) * HD + p) * DS + n] = acc;
  }
}

// sequential inter-chunk scan; prev[c] = carry before chunk c
__global__ void ssd_scan(const float* __restrict__ states, const float* __restrict__ cdec,
                         float* __restrict__ prev) {
  int idx = blockIdx.x * blockDim.x + threadIdx.x;
  if (idx >= BW * NH * HD * DS) return;
  int n = idx & 31;
  int p = (idx >> 5) & 31;
  int h = (idx >> 10) & 15;
  int b = idx >> 14;
  float carry = 0.0f;
  for (int cc = 0; cc < NC; ++cc) {
    size_t sidx = ((((size_t)b * NC + cc) * NH + h) * HD + p) * DS + n;
    prev[sidx] = carry;
    carry = carry * cdec[((size_t)b * NH + h) * NC + cc] + states[sidx];
  }
}

// per-(b,c,h): Yo from carried state + D skip term, accumulated into yb
__global__ void ssd_yoff(const float* __restrict__ xbcc, const float* __restrict__ prev,
                         const float* __restrict__ acs, const float* __restrict__ Dh,
                         float* __restrict__ yb) {
  int h = blockIdx.x % NH;
  int c = (blockIdx.x / NH) % NC;
  int b = blockIdx.x / (NH * NC);
  __shared__ float sP[HD][DS], sC[CH][DS], sA[CH];
  int tid = threadIdx.x;
  for (int i = tid; i < HD * DS; i += 256)
    sP[i >> 5][i & 31] = prev[((((size_t)b * NC + c) * NH + h) * HD + (i >> 5)) * DS + (i & 31)];
  for (int i = tid; i < CH * DS; i += 256) {
    int l = i >> 5, n = i & 31;
    sC[l][n] = xbcc[((size_t)b * T + c * CH + l) * CONVD + DI + DS + n];
  }
  for (int l = tid; l < CH; l += 256)
    sA[l] = acs[(((size_t)b * NH + h) * NC + c) * CH + l];
  __syncthreads();
  float dh = Dh[h];
  for (int i = tid; i < CH * HD; i += 256) {
    int l = i >> 5, p = i & 31;
    float acc = 0.0f;
    for (int n = 0; n < DS; ++n) acc += sC[l][n] * sP[p][n];
    acc *= expf(sA[l]);
    int t = c * CH + l;
    size_t yi = ((size_t)b * T + t) * DI + h * HD + p;
    float xv = xbcc[((size_t)b * T + t) * CONVD + h * HD + p];
    yb[yi] += acc + dh * xv;
  }
}

// g = y * silu(z); RMSNorm(512) * nw — in place on yb
__global__ void gate_rmsnorm(const float* __restrict__ zx, const float* __restrict__ nw,
                             float* __restrict__ yb) {
  int bt = blockIdx.x;
  int tid = threadIdx.x;
  __shared__ float red[256];
  size_t ybase = (size_t)bt * DI;
  size_t zbase = (size_t)bt * DIP;
  float g0 = yb[ybase + tid]       * siluf(zx[zbase + tid]);
  float g1 = yb[ybase + tid + 256] * siluf(zx[zbase + tid + 256]);
  red[tid] = g0 * g0 + g1 * g1;
  __syncthreads();
  for (int st = 128; st > 0; st >>= 1) {
    if (tid < st) red[tid] += red[tid + st];
    __syncthreads();
  }
  float scale = rsqrtf(red[0] / (float)DI + EPS);
  yb[ybase + tid]       = g0 * scale * nw[tid];
  yb[ybase + tid + 256] = g1 * scale * nw[tid + 256];
}

// x = LayerNorm(add + x) over 256, blockDim=256, one row per block
__global__ void add_layernorm(const float* __restrict__ add, float* __restrict__ x,
                              const float* __restrict__ w, const float* __restrict__ b) {
  int row = blockIdx.x;
  int d = threadIdx.x;
  __shared__ float r1[256], r2[256];
  size_t base = (size_t)row * DM;
  float v = add[base + d] + x[base + d];
  r1[d] = v;
  r2[d] = v * v;
  __syncthreads();
  for (int st = 128; st > 0; st >>= 1) {
    if (d < st) { r1[d] += r1[d + st]; r2[d] += r2[d + st]; }
    __syncthreads();
  }
  float m   = r1[0] / (float)DM;
  float var = r2[0] / (float)DM - m * m;
  x[base + d] = (v - m) * rsqrtf(var + EPS) * w[d] + b[d];
}

// ======================================================================
// Aggregation head
// ======================================================================
__global__ void feats_mean(const float* __restrict__ x, float* __restrict__ feats) {
  int idx = blockIdx.x * blockDim.x + threadIdx.x;
  if (idx >= BW * DM) return;
  int d = idx & (DM - 1);
  int bw = idx >> 8;
  float s = 0.0f;
  for (int t = 0; t < T; ++t) s += x[((size_t)bw * T + t) * DM + d];
  feats[idx] = s * (1.0f / (float)T);
}

__global__ void mask_kernel(const float* __restrict__ win, float* __restrict__ maskf) {
  int bw = blockIdx.x;
  int tid = threadIdx.x;
  __shared__ float red[256];
  float s = 0.0f;
  for (int i = tid; i < CIN * T; i += 256) s += win[(size_t)bw * CIN * T + i];
  red[tid] = s;
  __syncthreads();
  for (int st = 128; st > 0; st >>= 1) {
    if (tid < st) red[tid] += red[tid + st];
    __syncthreads();
  }
  if (tid == 0) maskf[bw] = (red[0] != 0.0f) ? 1.0f : 0.0f;
}

// 8-head attention over 16 windows; thread = head*16 + query
__global__ void mha_attn(const float* __restrict__ qkv, const float* __restrict__ maskf,
                         float* __restrict__ att) {
  int b = blockIdx.x;
  int tid = threadIdx.x;        // 0..127
  int head = tid >> 4;
  int q = tid & 15;
  const float scale = 0.1767766952966369f; // 1/sqrt(32)
  const float* qv = qkv + ((size_t)(b * NWIN + q) * 768) + head * 32;
  float s[NWIN];
  float mx = -1e30f;
  for (int k = 0; k < NWIN; ++k) {
    const float* kv = qkv + ((size_t)(b * NWIN + k) * 768) + 256 + head * 32;
    float d = 0.0f;
    for (int j = 0; j < 32; ++j) d += qv[j] * kv[j];
    d *= scale;
    if (maskf[b * NWIN + k] == 0.0f) d = -1e9f;
    s[k] = d;
    mx = fmaxf(mx, d);
  }
  float ssum = 0.0f;
  for (int k = 0; k < NWIN; ++k) { s[k] = expf(s[k] - mx); ssum += s[k]; }
  float inv = 1.0f / ssum;
  for (int j = 0; j < 32; ++j) {
    float acc = 0.0f;
    for (int k = 0; k < NWIN; ++k)
      acc += s[k] * qkv[((size_t)(b * NWIN + k) * 768) + 512 + head * 32 + j];
    att[((size_t)(b * NWIN + q)) * DM + head * 32 + j] = acc * inv;
  }
}

// masked mean pool + 3-layer MLP head; one block per subject
__global__ void head_mlp(const float* __restrict__ h, const float* __restrict__ maskf,
                         const float* __restrict__ w1, const float* __restrict__ b1,
                         const float* __restrict__ w2, const float* __restrict__ b2,
                         const float* __restrict__ w3, const float* __restrict__ b3,
                         float* __restrict__ out) {
  int b = blockIdx.x;
  int tid = threadIdx.x;        // 128 threads
  __shared__ float subj[DM], r1[128], r2[64], msum;
  if (tid == 0) {
    float s = 0.0f;
    for (int w = 0; w < NWIN; ++w) s += maskf[b * NWIN + w];
    msum = fmaxf(s, 1.0f);
  }
  __syncthreads();
  for (int d = tid; d < DM; d += 128) {
    float s = 0.0f;
    for (int w = 0; w < NWIN; ++w)
      s += h[((size_t)(b * NWIN + w)) * DM + d] * maskf[b * NWIN + w];
    subj[d] = s / msum;
  }
  __syncthreads();
  {
    float acc = b1[tid];
    for (int k = 0; k < DM; ++k) acc += subj[k] * w1[(size_t)tid * DM + k];
    r1[tid] = geluf(acc);
  }
  __syncthreads();
  if (tid < 64) {
    float acc = b2[tid];
    for (int k = 0; k < 128; ++k) acc += r1[k] * w2[(size_t)tid * 128 + k];
    r2[tid] = geluf(acc);
  }
  __syncthreads();
  if (tid == 0) {
    float acc = b3[0];
    for (int k = 0; k < 64; ++k) acc += r2[k] * w3[k];
    out[b] = acc;
  }
}

// ======================================================================
// Host orchestration
// ======================================================================
static inline void launch_gemm(const float* A, const float* W, const float* bias,
                               float* C, int M, int N, int K, hipStream_t s) {
  int ngroups = ((N / 16) + 3) / 4;
  int waves = (M / 16) * ngroups;
  int blocks = (waves + 3) / 4;               // 4 waves / block (128 thr)
  hipLaunchKernelGGL(gemm_nt_wmma, dim3(blocks), dim3(128), 0, s, A, W, bias, C, M, N, K);
}

extern "C" void kernel_launch(void* const* d_in, const int* in_sizes, int n_in,
                              void* d_out, int out_size, void* d_ws, size_t ws_size,
                              hipStream_t stream) {
  (void)in_sizes; (void)n_in; (void)out_size; (void)ws_size;
  const float* windows   = (const float*)d_in[0];
  const float* conv1_w   = (const float*)d_in[1];
  const float* conv1_b   = (const float*)d_in[2];
  const float* bn_g      = (const float*)d_in[3];
  const float* bn_b      = (const float*)d_in[4];
  const float* bn_rm     = (const float*)d_in[5];
  const float* bn_rv     = (const float*)d_in[6];
  const float* conv2_w   = (const float*)d_in[7];
  const float* conv2_b   = (const float*)d_in[8];
  const float* in_proj_w = (const float*)d_in[9];
  const float* mconv_w   = (const float*)d_in[10];
  const float* mconv_b   = (const float*)d_in[11];
  const float* dt_bias   = (const float*)d_in[12];
  const float* A_log     = (const float*)d_in[13];
  const float* D_h       = (const float*)d_in[14];
  const float* mnorm_w   = (const float*)d_in[15];
  const float* out_proj_w= (const float*)d_in[16];
  const float* ln_w      = (const float*)d_in[17];
  const float* ln_b      = (const float*)d_in[18];
  const float* mha_in_w  = (const float*)d_in[19];
  const float* mha_in_b  = (const float*)d_in[20];
  const float* mha_out_w = (const float*)d_in[21];
  const float* mha_out_b = (const float*)d_in[22];
  const float* agg_ln_w  = (const float*)d_in[23];
  const float* agg_ln_b  = (const float*)d_in[24];
  const float* h1_w      = (const float*)d_in[25];
  const float* h1_b      = (const float*)d_in[26];
  const float* h2_w      = (const float*)d_in[27];
  const float* h2_b      = (const float*)d_in[28];
  const float* h3_w      = (const float*)d_in[29];
  const float* h3_b      = (const float*)d_in[30];
  float* out = (float*)d_out;

  // workspace carve-up (floats)
  float* ws = (float*)d_ws;
  size_t off = 0;
  float* A1     = ws + off; off += (size_t)BT * 128;        // conv1 out (bt,128)
  float* X      = ws + off; off += (size_t)BT * DM;         // activations (bt,256)
  float* ZX     = ws + off; off += (size_t)BT * DIP;        // in_proj out
  float* XBCC   = ws + off; off += (size_t)BT * CONVD;      // conv+silu xBC
  float* DT     = ws + off; off += (size_t)BT * NH;
  float* ACS    = ws + off; off += (size_t)BW * NH * NC * CH;
  float* CDEC   = ws + off; off += (size_t)BW * NH * NC;
  float* YB     = ws + off; off += (size_t)BT * DI;         // ssd out / gated g
  float* STATES = ws + off; off += (size_t)BW * NC * NH * HD * DS;
  float* PREV   = ws + off; off += (size_t)BW * NC * NH * HD * DS;
  float* Y2     = ws + off; off += (size_t)BT * DM;         // mamba/mha proj out
  float* FEATS  = ws + off; off += (size_t)BW * DM;
  float* QKV    = ws + off; off += (size_t)BW * 768;
  float* ATT    = ws + off; off += (size_t)BW * DM;
  float* MASKF  = ws + off; off += (size_t)BW;

  // ---- front-end ----
  hipLaunchKernelGGL(conv1_bn_gelu, dim3((BW * 128 * T + 255) / 256), dim3(256), 0, stream,
                     windows, conv1_w, conv1_b, bn_g, bn_b, bn_rm, bn_rv, A1);
  launch_gemm(A1, conv2_w, conv2_b, X, BT, DM, 128, stream);   // -> (bt,256)

  // ---- 6 Mamba2 layers ----
  for (int i = 0; i < NLAYERS; ++i) {
    launch_gemm(X, in_proj_w + (size_t)i * DIP * DM, nullptr, ZX, BT, DIP, DM, stream);
    hipLaunchKernelGGL(dwconv_silu, dim3((BT * CONVD + 255) / 256), dim3(256), 0, stream,
                       ZX, mconv_w + (size_t)i * CONVD * 4, mconv_b + (size_t)i * CONVD, XBCC);
    hipLaunchKernelGGL(dt_softplus, dim3((BT * NH + 255) / 256), dim3(256), 0, stream,
                       ZX, dt_bias + (size_t)i * NH, DT);
    hipLaunchKernelGGL(ssd_cumsum, dim3((BW * NH * NC + 255) / 256), dim3(256), 0, stream,
                       DT, A_log + (size_t)i * NH, ACS, CDEC);
    hipLaunchKernelGGL(ssd_chunk, dim3(BW * NC * NH), dim3(256), 0, stream,
                       XBCC, DT, ACS, YB, STATES);
    hipLaunchKernelGGL(ssd_scan, dim3((BW * NH * HD * DS + 255) / 256), dim3(256), 0, stream,
                       STATES, CDEC, PREV);
    hipLaunchKernelGGL(ssd_yoff, dim3(BW * NC * NH), dim3(256), 0, stream,
                       XBCC, PREV, ACS, D_h + (size_t)i * NH, YB);
    hipLaunchKernelGGL(gate_rmsnorm, dim3(BT), dim3(256), 0, stream,
                       ZX, mnorm_w + (size_t)i * DI, YB);
    launch_gemm(YB, out_proj_w + (size_t)i * DM * DI, nullptr, Y2, BT, DM, DI, stream);
    hipLaunchKernelGGL(add_layernorm, dim3(BT), dim3(256), 0, stream,
                       Y2, X, ln_w + (size_t)i * DM, ln_b + (size_t)i * DM);
  }

  // ---- aggregation ----
  hipLaunchKernelGGL(feats_mean, dim3((BW * DM + 255) / 256), dim3(256), 0, stream, X, FEATS);
  hipLaunchKernelGGL(mask_kernel, dim3(BW), dim3(256), 0, stream, windows, MASKF);

  for (int j = 0; j < 2; ++j) {
    launch_gemm(FEATS, mha_in_w + (size_t)j * 768 * DM, mha_in_b + (size_t)j * 768,
                QKV, BW, 768, DM, stream);
    hipLaunchKernelGGL(mha_attn, dim3(BATCH), dim3(128), 0, stream, QKV, MASKF, ATT);
    launch_gemm(ATT, mha_out_w + (size_t)j * DM * DM, mha_out_b + (size_t)j * DM,
                Y2, BW, DM, DM, stream);
    hipLaunchKernelGGL(add_layernorm, dim3(BW), dim3(256), 0, stream,
                       Y2, FEATS, agg_ln_w + (size_t)j * DM, agg_ln_b + (size_t)j * DM);
  }

  hipLaunchKernelGGL(head_mlp, dim3(BATCH), dim3(128), 0, stream,
                     FEATS, MASKF, h1_w, h1_b, h2_w, h2_b, h3_w, h3_b, out);
}